// LossTGMVector_83227876262128
// MI455X (gfx1250) — compile-verified
//
#include <hip/hip_runtime.h>
#include <math.h>

// Problem constants (fixed by the reference)
#define BB 2
#define TT 32
#define HH 518
#define WW 518
#define HW (HH * WW)          // 268324
#define HW4 (HW / 4)          // 67081 (exact)
#define THW (TT * HW)         // 8586368
#define THW4 (THW / 4)        // 2146592 (exact)
#define NROWS (BB * (TT - 1)) // 62
#define EPSF 1e-6f
#define STATIC_THF 0.05f
#define Q_TRIM 0.8f           // 1.0 - TRIM_RATIO

typedef __attribute__((ext_vector_type(2))) float v2f;
typedef __attribute__((ext_vector_type(8))) float v8f;

// -------------------------------------------------------------------------
// Wave32 full reduction using V_WMMA_F32_16X16X4_F32 (codegen-verified in
// round 1: emits v_wmma_f32_16x16x4_f32).
// A (16x4 f32, 2 VGPRs/lane): a.x = lane value, a.y = 0.
// B = all ones. D[m][*] = v_m + v_{m+16}; summing the 8 D VGPRs gives the
// half-wave sum, one xor-16 shuffle completes the 32-lane sum on all lanes.
// EXEC must be all ones at the call site.
// -------------------------------------------------------------------------
__device__ __forceinline__ float wave_reduce_wmma(float v) {
  v2f a; a.x = v;    a.y = 0.0f;
  v2f b; b.x = 1.0f; b.y = 1.0f;
  v8f c = {0.f, 0.f, 0.f, 0.f, 0.f, 0.f, 0.f, 0.f};
  v8f d = __builtin_amdgcn_wmma_f32_16x16x4_f32(
      /*neg_a=*/false, a, /*neg_b=*/false, b,
      /*c_mod=*/(short)0, c, /*reuse_a=*/false, /*reuse_b=*/false);
  float s = d[0] + d[1] + d[2] + d[3] + d[4] + d[5] + d[6] + d[7];
  s += __shfl_xor(s, 16, 32);
  return s;
}

// -------------------------------------------------------------------------
// Workspace zero-init (accumulators only; err array is fully overwritten)
// -------------------------------------------------------------------------
__global__ void k_init(double* alignSums, int* nv, float* rowsum, float* rowcnt) {
  int t = threadIdx.x;
  if (t < BB * 5) alignSums[t] = 0.0;
  if (t < NROWS) { nv[t] = 0; rowsum[t] = 0.0f; rowcnt[t] = 0.0f; }
}

// -------------------------------------------------------------------------
// Kernel 1: per-batch masked sums for least-squares alignment.
// 128-bit loads: float4 pred/gt + int4 mask per iteration.
//   sums[b][0..4] = {count, S(raw), S(gt_disp), S(raw*gt_disp), S(raw^2)}
// -------------------------------------------------------------------------
__global__ void k_align_sums(const float* __restrict__ pred,
                             const float* __restrict__ gt,
                             const int* __restrict__ mask,
                             double* __restrict__ sums) {
  const int b = blockIdx.y;
  const float4* __restrict__ p4 = (const float4*)(pred + (size_t)b * THW);
  const float4* __restrict__ g4 = (const float4*)(gt   + (size_t)b * THW);
  const int4*   __restrict__ m4 = (const int4*)(mask + (size_t)b * THW);

  float c = 0.f, sr = 0.f, sg = 0.f, srg = 0.f, srr = 0.f;
  const size_t stride = (size_t)gridDim.x * blockDim.x;
  for (size_t i = (size_t)blockIdx.x * blockDim.x + threadIdx.x; i < (size_t)THW4; i += stride) {
    float4 p = p4[i];
    float4 g = g4[i];
    int4   m = m4[i];
    float pe[4] = {p.x, p.y, p.z, p.w};
    float ge[4] = {g.x, g.y, g.z, g.w};
    int   me[4] = {m.x, m.y, m.z, m.w};
#pragma unroll
    for (int j = 0; j < 4; ++j) {
      float mm  = (me[j] != 0) ? 1.0f : 0.0f;
      float raw = fmaxf(pe[j], EPSF);
      float gd  = 1.0f / fmaxf(ge[j], EPSF);
      c   += mm;
      sr  += mm * raw;
      sg  += mm * gd;
      srg += mm * raw * gd;
      srr += mm * raw * raw;
    }
  }
  float vals[5] = {c, sr, sg, srg, srr};
  __shared__ float red[8][5];
  const int wave = threadIdx.x >> 5, lane = threadIdx.x & 31;
#pragma unroll
  for (int j = 0; j < 5; ++j) {
    float w = wave_reduce_wmma(vals[j]);     // EXEC all ones here
    if (lane == 0) red[wave][j] = w;
  }
  __syncthreads();
  if (threadIdx.x == 0) {
#pragma unroll
    for (int j = 0; j < 5; ++j) {
      double acc = 0.0;
      for (int w = 0; w < 8; ++w) acc += (double)red[w][j];
      unsafeAtomicAdd(&sums[b * 5 + j], acc);   // global_atomic_add_f64
    }
  }
}

// -------------------------------------------------------------------------
// Kernel 2: solve per-batch scale/shift (tiny)
// -------------------------------------------------------------------------
__global__ void k_solve(const double* __restrict__ sums,
                        float* __restrict__ sArr, float* __restrict__ tArr) {
  int b = threadIdx.x;
  if (b < BB) {
    double cnt = sums[b * 5 + 0];
    double c2 = (cnt < 1.0) ? 1.0 : cnt;
    double mr = sums[b * 5 + 1] / c2;
    double mg = sums[b * 5 + 2] / c2;
    double cov = sums[b * 5 + 3] - c2 * mr * mg;
    double var = sums[b * 5 + 4] - c2 * mr * mr;
    if (var < 1e-6) var = 1e-6;
    double s = cov / var;
    sArr[b] = (float)s;
    tArr[b] = (float)(mg - s * mr);
  }
}

// -------------------------------------------------------------------------
// Kernel 3: build err array, 4 pixels per thread (B128 loads/stores).
// Non-static pixels encoded as +inf so they sort to the end (bitwise
// monotone with the reference's "invalid -> +inf" nanquantile trick).
// Also counts static pixels per row (nv).
// -------------------------------------------------------------------------
__global__ void k_err(const float* __restrict__ pred,
                      const float* __restrict__ gt,
                      const int* __restrict__ mask,
                      const float* __restrict__ sArr,
                      const float* __restrict__ tArr,
                      float* __restrict__ err,
                      int* __restrict__ nv) {
  const int n = blockIdx.y;
  const int b = n / (TT - 1);
  const int f = n % (TT - 1);
  const int q4 = blockIdx.x * blockDim.x + threadIdx.x;
  __shared__ int cnt;
  if (threadIdx.x == 0) cnt = 0;
  __syncthreads();
  int nStatic = 0;
  if (q4 < HW4) {
    const size_t r0 = ((size_t)(b * TT + f)) * HW4 + q4;   // frame f, float4 idx
    const size_t r1 = r0 + HW4;                            // frame f+1
    float4 p0 = ((const float4*)pred)[r0];
    float4 p1 = ((const float4*)pred)[r1];
    float4 g0 = ((const float4*)gt)[r0];
    float4 g1 = ((const float4*)gt)[r1];
    int4   m0 = ((const int4*)mask)[r0];
    int4   m1 = ((const int4*)mask)[r1];
    const float s = sArr[b], t = tArr[b];

    float p0e[4] = {p0.x, p0.y, p0.z, p0.w};
    float p1e[4] = {p1.x, p1.y, p1.z, p1.w};
    float g0e[4] = {g0.x, g0.y, g0.z, g0.w};
    float g1e[4] = {g1.x, g1.y, g1.z, g1.w};
    int   m0e[4] = {m0.x, m0.y, m0.z, m0.w};
    int   m1e[4] = {m1.x, m1.y, m1.z, m1.w};
    float ee[4];
#pragma unroll
    for (int j = 0; j < 4; ++j) {
      float a0 = fmaxf(p0e[j], EPSF) * s + t;
      float a1 = fmaxf(p1e[j], EPSF) * s + t;
      float gd0 = 1.0f / fmaxf(g0e[j], EPSF);
      float gd1 = 1.0f / fmaxf(g1e[j], EPSF);
      float dd = fabsf(a1 - a0);
      float gg = fabsf(gd1 - gd0);
      bool st = (m0e[j] != 0) && (m1e[j] != 0) && (fabsf(g1e[j] - g0e[j]) < STATIC_THF);
      ee[j] = st ? fabsf(dd - gg) : __uint_as_float(0x7F800000u); // +inf sentinel
      nStatic += st ? 1 : 0;
    }
    float4 ev = make_float4(ee[0], ee[1], ee[2], ee[3]);
    ((float4*)err)[(size_t)n * HW4 + q4] = ev;
  }
  atomicAdd(&cnt, nStatic);
  __syncthreads();
  if (threadIdx.x == 0 && cnt > 0) atomicAdd(&nv[n], cnt);
}

// -------------------------------------------------------------------------
// Kernel 4: exact dual radix-select (byte digits, MSB->LSB) per row.
// uint4 (B128) loads; the 66.5 MB err array is L2-resident so the 4 scans
// run at L2 bandwidth. Selects both klo and khi order statistics in one
// set of scans via dual LDS histograms, then lerps exactly like the
// reference's linear-interpolation nanquantile.
// -------------------------------------------------------------------------
__global__ void k_select(const float* __restrict__ err,
                         const int* __restrict__ nv,
                         float* __restrict__ thresh) {
  const int n = blockIdx.x;
  const uint4* __restrict__ row4 = (const uint4*)(err + (size_t)n * HW);
  __shared__ int hist0[256], hist1[256];
  __shared__ unsigned sh_pfx0, sh_pfx1;
  __shared__ int sh_k0, sh_k1;
  const int tid = threadIdx.x;

  const int nvr = nv[n];
  const float pos = Q_TRIM * (float)((nvr - 1 > 0) ? (nvr - 1) : 0);
  const int klo = (int)floorf(pos);
  const int khi = (int)ceilf(pos);
  const float frac = pos - (float)klo;

  if (tid == 0) { sh_pfx0 = 0u; sh_pfx1 = 0u; sh_k0 = klo; sh_k1 = khi; }

  for (int pass = 3; pass >= 0; --pass) {
    for (int i = tid; i < 256; i += blockDim.x) { hist0[i] = 0; hist1[i] = 0; }
    __syncthreads();
    const unsigned pfx0 = sh_pfx0, pfx1 = sh_pfx1;
    const int shift = 8 * pass;
    for (int q4 = tid; q4 < HW4; q4 += blockDim.x) {
      uint4 uv = row4[q4];
      unsigned ue[4] = {uv.x, uv.y, uv.z, uv.w};
#pragma unroll
      for (int j = 0; j < 4; ++j) {
        unsigned u = ue[j];
        unsigned hi = (pass == 3) ? 0u : (u >> (shift + 8));
        unsigned bin = (u >> shift) & 255u;
        if (hi == pfx0) atomicAdd(&hist0[bin], 1);
        if (hi == pfx1) atomicAdd(&hist1[bin], 1);
      }
    }
    __syncthreads();
    if (tid == 0) {
      int k = sh_k0, cum = 0, bsel = 0;
      for (; bsel < 256; ++bsel) { int h = hist0[bsel]; if (cum + h > k) break; cum += h; }
      sh_pfx0 = (sh_pfx0 << 8) | (unsigned)bsel; sh_k0 = k - cum;
      k = sh_k1; cum = 0; bsel = 0;
      for (; bsel < 256; ++bsel) { int h = hist1[bsel]; if (cum + h > k) break; cum += h; }
      sh_pfx1 = (sh_pfx1 << 8) | (unsigned)bsel; sh_k1 = k - cum;
    }
    __syncthreads();
  }

  if (tid == 0) {
    float vlo = __uint_as_float(sh_pfx0);
    float vhi = __uint_as_float(sh_pfx1);
    float th = vlo * (1.0f - frac) + vhi * frac;
    thresh[n] = (nvr > 0) ? th : __uint_as_float(0x7FC00000u); // NaN -> keep none
  }
}

// -------------------------------------------------------------------------
// Kernel 5: trimmed masked mean per row (B128 loads). +inf sentinels / NaN
// thresholds fail `e <= th`, reproducing keep = static & (err<=thresh).
// WMMA wave reduction, then f32 atomics into rowsum/rowcnt.
// -------------------------------------------------------------------------
__global__ void k_trim(const float* __restrict__ err,
                       const float* __restrict__ thresh,
                       float* __restrict__ rowsum,
                       float* __restrict__ rowcnt) {
  const int n = blockIdx.y;
  const float th = thresh[n];
  const float4* __restrict__ row4 = (const float4*)(err + (size_t)n * HW);
  float s = 0.f, c = 0.f;
  const int stride = gridDim.x * blockDim.x;
  for (int q4 = blockIdx.x * blockDim.x + threadIdx.x; q4 < HW4; q4 += stride) {
    float4 ev = row4[q4];
    float ee[4] = {ev.x, ev.y, ev.z, ev.w};
#pragma unroll
    for (int j = 0; j < 4; ++j) {
      if (ee[j] <= th) { s += ee[j]; c += 1.0f; }
    }
  }
  __shared__ float red[8][2];
  const int wave = threadIdx.x >> 5, lane = threadIdx.x & 31;
  float sw = wave_reduce_wmma(s);
  float cw = wave_reduce_wmma(c);
  if (lane == 0) { red[wave][0] = sw; red[wave][1] = cw; }
  __syncthreads();
  if (threadIdx.x == 0) {
    float ts = 0.f, tc = 0.f;
    for (int w = 0; w < 8; ++w) { ts += red[w][0]; tc += red[w][1]; }
    unsafeAtomicAdd(&rowsum[n], ts);
    unsafeAtomicAdd(&rowcnt[n], tc);
  }
}

// -------------------------------------------------------------------------
// Kernel 6: loss_frame = rowsum / max(rowcnt,1); out = mean over 62 rows
// -------------------------------------------------------------------------
__global__ void k_final(const float* __restrict__ rowsum,
                        const float* __restrict__ rowcnt,
                        float* __restrict__ out) {
  __shared__ float buf[64];
  const int t = threadIdx.x;
  float v = 0.f;
  if (t < NROWS) {
    float c = rowcnt[t];
    if (c < 1.0f) c = 1.0f;
    v = rowsum[t] / c;
  }
  buf[t] = v;
  __syncthreads();
  for (int s = 32; s > 0; s >>= 1) {
    if (t < s) buf[t] += buf[t + s];
    __syncthreads();
  }
  if (t == 0) out[0] = buf[0] / (float)NROWS;
}

// -------------------------------------------------------------------------
// Host launch
// -------------------------------------------------------------------------
extern "C" void kernel_launch(void* const* d_in, const int* in_sizes, int n_in,
                              void* d_out, int out_size, void* d_ws, size_t ws_size,
                              hipStream_t stream) {
  const float* pred = (const float*)d_in[0];
  const float* gt   = (const float*)d_in[1];
  const int*   mask = (const int*)d_in[2];
  float* out = (float*)d_out;

  char* ws = (char*)d_ws;
  double* alignSums = (double*)(ws + 0);      // BB*5 doubles   (80 B)
  float*  sArr      = (float*)(ws + 128);     // BB floats
  float*  tArr      = (float*)(ws + 160);     // BB floats
  int*    nv        = (int*)(ws + 256);       // NROWS ints
  float*  thresh    = (float*)(ws + 768);     // NROWS floats
  float*  rowsum    = (float*)(ws + 1280);    // NROWS floats
  float*  rowcnt    = (float*)(ws + 1792);    // NROWS floats
  float*  err       = (float*)(ws + 4096);    // NROWS*HW floats (~66.5 MB), 16B-aligned rows

  const int qBlocks4 = (HW4 + 255) / 256;     // 263

  k_init<<<1, 64, 0, stream>>>(alignSums, nv, rowsum, rowcnt);
  k_align_sums<<<dim3(1024, BB), 256, 0, stream>>>(pred, gt, mask, alignSums);
  k_solve<<<1, 32, 0, stream>>>(alignSums, sArr, tArr);
  k_err<<<dim3(qBlocks4, NROWS), 256, 0, stream>>>(pred, gt, mask, sArr, tArr, err, nv);
  k_select<<<NROWS, 256, 0, stream>>>(err, nv, thresh);
  k_trim<<<dim3(qBlocks4, NROWS), 256, 0, stream>>>(err, thresh, rowsum, rowcnt);
  k_final<<<1, 64, 0, stream>>>(rowsum, rowcnt, out);
}